// GCNEmb_47356309406258
// MI455X (gfx1250) — compile-verified
//
#include <hip/hip_runtime.h>

typedef __attribute__((ext_vector_type(2))) float v2f;
typedef __attribute__((ext_vector_type(8))) float v8f;

// ---------------------------------------------------------------------------
// utility kernels
// ---------------------------------------------------------------------------
__global__ void k_zero_f32(float* __restrict__ p, long long n) {
  long long i = (long long)blockIdx.x * blockDim.x + threadIdx.x;
  if (i < n) p[i] = 0.0f;
}

__global__ void k_degree(const int* __restrict__ dst, int E, unsigned* __restrict__ deg) {
  int e = blockIdx.x * blockDim.x + threadIdx.x;
  if (e < E) atomicAdd(&deg[dst[e]], 1u);
}

// deg buffer (uint counts) -> dis = rsqrt(deg + 1)   (the +1 is the self loop)
__global__ void k_dis(float* __restrict__ dis, int N) {
  int i = blockIdx.x * blockDim.x + threadIdx.x;
  if (i < N) {
    unsigned c = ((const unsigned*)dis)[i];
    dis[i] = rsqrtf((float)c + 1.0f);
  }
}

// ---------------------------------------------------------------------------
// Y = X @ W  via V_WMMA_F32_16X16X4_F32, one wave per 16x16 output tile.
// X: [M,K] row-major, W: [K,Nout] row-major, Y: [M,Nout] row-major.
// M % 16 == 0, K % 4 == 0, Nout % 16 == 0 (true for this workload).
// ---------------------------------------------------------------------------
__global__ void k_gemm_wmma(const float* __restrict__ X, const float* __restrict__ W,
                            float* __restrict__ Y, int M, int K, int Nout) {
  const int lane = threadIdx.x & 31;
  const int wave = threadIdx.x >> 5;
  const int Nt = Nout >> 4;
  const int gw = blockIdx.x * (blockDim.x >> 5) + wave;
  const int totalTiles = (M >> 4) * Nt;
  if (gw >= totalTiles) return;                 // wave-uniform: EXEC stays all-ones
  const int mt = gw / Nt;
  const int nt = gw - mt * Nt;
  const int m0 = mt << 4;
  const int n0 = nt << 4;

  const int half = lane >> 4;                   // 0: K pair {0,1}, 1: K pair {2,3}
  const int l15  = lane & 15;

  // A layout (16x4 fp32): lanes 0-15 -> M=l15, v0=K0,v1=K1; lanes 16-31 -> K2,K3
  const float* Arow = X + (long long)(m0 + l15) * K + (half << 1);
  // B layout (4x16 fp32): lanes 0-15 -> K row (k0|k0+1), N=n0+l15; lanes 16-31 -> +2
  const float* Bcol = W + (long long)(half << 1) * Nout + (n0 + l15);

  v8f acc = {};
#pragma unroll 4
  for (int k = 0; k < K; k += 4) {
    v2f a = *(const v2f*)(Arow + k);            // K,K+1 contiguous, 8B aligned
    v2f b;
    b.x = Bcol[(long long)k * Nout];
    b.y = Bcol[(long long)k * Nout + Nout];
    acc = __builtin_amdgcn_wmma_f32_16x16x4_f32(
        /*neg_a=*/false, a, /*neg_b=*/false, b,
        /*c_mod=*/(short)0, acc, /*reuse_a=*/false, /*reuse_b=*/false);
  }

  // C/D layout: VGPR v -> row m0+v (lanes 0-15) / m0+8+v (lanes 16-31), col n0+l15
  float* Yo = Y + (long long)(m0 + (half << 3)) * Nout + (n0 + l15);
#pragma unroll
  for (int v = 0; v < 8; ++v) Yo[(long long)v * Nout] = acc[v];
}

// ---------------------------------------------------------------------------
// agg[dst] += Y[src] * (dis[src]*dis[dst]) over all edges.
// One thread per (edge, 4-float chunk); csh = log2(D/4).
// ---------------------------------------------------------------------------
__global__ void k_scatter(const int* __restrict__ src, const int* __restrict__ dst,
                          const float* __restrict__ dis, const float* __restrict__ Y,
                          float* __restrict__ agg, int E, int D, int csh) {
  long long t = (long long)blockIdx.x * blockDim.x + threadIdx.x;
  long long total = (long long)E << csh;
  if (t >= total) return;
  int e = (int)(t >> csh);
  int c = (int)(t & ((1 << csh) - 1));
  int s = src[e];
  int dn = dst[e];
  float norm = dis[s] * dis[dn];
  const float4 y = *(const float4*)(Y + (long long)s * D + (c << 2));
  float* o = agg + (long long)dn * D + (c << 2);
  unsafeAtomicAdd(o + 0, y.x * norm);           // hardware global_atomic_add_f32
  unsafeAtomicAdd(o + 1, y.y * norm);
  unsafeAtomicAdd(o + 2, y.z * norm);
  unsafeAtomicAdd(o + 3, y.w * norm);
}

// out = agg + Y*dis^2 (self loop) + bias, optional ReLU.  dsh = log2(D).
__global__ void k_finalize(const float* __restrict__ agg, const float* __restrict__ Y,
                           const float* __restrict__ dis, const float* __restrict__ b,
                           float* __restrict__ out, long long total, int dsh, int relu) {
  long long t = (long long)blockIdx.x * blockDim.x + threadIdx.x;
  if (t >= total) return;
  int row = (int)(t >> dsh);
  int col = (int)(t & ((1 << dsh) - 1));
  float s = dis[row];
  float v = agg[t] + Y[t] * (s * s) + b[col];
  out[t] = relu ? fmaxf(v, 0.0f) : v;
}

// ---------------------------------------------------------------------------
extern "C" void kernel_launch(void* const* d_in, const int* in_sizes, int n_in,
                              void* d_out, int out_size, void* d_ws, size_t ws_size,
                              hipStream_t stream) {
  const float* feat = (const float*)d_in[0];
  const int*   ei   = (const int*)d_in[1];
  const float* W0   = (const float*)d_in[2];
  const float* b0   = (const float*)d_in[3];
  const float* W1   = (const float*)d_in[4];
  const float* b1   = (const float*)d_in[5];
  const float* W2   = (const float*)d_in[6];
  const float* b2   = (const float*)d_in[7];

  const int N = in_sizes[0] / 128;   // 100000
  const int E = in_sizes[1] / 2;     // 1600000
  const int* src = ei;
  const int* dst = ei + E;

  float* ws  = (float*)d_ws;
  float* dis = ws;                         // N floats (doubles as u32 degree counts)
  float* Y   = ws + (size_t)N;             // N x 64
  float* AGG = ws + (size_t)N * 65;        // N x 64
  float* X   = ws + (size_t)N * 129;       // N x 64

  const int TB = 256;
  auto nb = [](long long n, int tb) { return (unsigned)((n + tb - 1) / tb); };

  // ---- degree / symmetric normalization -----------------------------------
  k_zero_f32<<<nb(N, TB), TB, 0, stream>>>(dis, N);
  k_degree<<<nb(E, TB), TB, 0, stream>>>(dst, E, (unsigned*)dis);
  k_dis<<<nb(N, TB), TB, 0, stream>>>(dis, N);

  // ---- layer 0: 128 -> 64, ReLU -------------------------------------------
  {
    const int K = 128, D = 64;
    const long long tiles = (long long)(N / 16) * (D / 16);
    k_gemm_wmma<<<nb(tiles * 32, TB), TB, 0, stream>>>(feat, W0, Y, N, K, D);
    k_zero_f32<<<nb((long long)N * D, TB), TB, 0, stream>>>(AGG, (long long)N * D);
    k_scatter<<<nb((long long)E * (D / 4), TB), TB, 0, stream>>>(src, dst, dis, Y, AGG, E, D, 4);
    k_finalize<<<nb((long long)N * D, TB), TB, 0, stream>>>(AGG, Y, dis, b0, X, (long long)N * D, 6, 1);
  }

  // ---- layer 1: 64 -> 64, ReLU --------------------------------------------
  {
    const int K = 64, D = 64;
    const long long tiles = (long long)(N / 16) * (D / 16);
    k_gemm_wmma<<<nb(tiles * 32, TB), TB, 0, stream>>>(X, W1, Y, N, K, D);
    k_zero_f32<<<nb((long long)N * D, TB), TB, 0, stream>>>(AGG, (long long)N * D);
    k_scatter<<<nb((long long)E * (D / 4), TB), TB, 0, stream>>>(src, dst, dis, Y, AGG, E, D, 4);
    k_finalize<<<nb((long long)N * D, TB), TB, 0, stream>>>(AGG, Y, dis, b1, X, (long long)N * D, 6, 1);
  }

  // ---- layer 2: 64 -> 32, no ReLU, write d_out ----------------------------
  {
    const int K = 64, D = 32;
    const long long tiles = (long long)(N / 16) * (D / 16);
    k_gemm_wmma<<<nb(tiles * 32, TB), TB, 0, stream>>>(X, W2, Y, N, K, D);
    k_zero_f32<<<nb((long long)N * D, TB), TB, 0, stream>>>(AGG, (long long)N * D);
    k_scatter<<<nb((long long)E * (D / 4), TB), TB, 0, stream>>>(src, dst, dis, Y, AGG, E, D, 3);
    k_finalize<<<nb((long long)N * D, TB), TB, 0, stream>>>(AGG, Y, dis, b2, (float*)d_out,
                                                            (long long)N * D, 5, 0);
  }
}